// GCN_31722628448491
// MI455X (gfx1250) — compile-verified
//
#include <hip/hip_runtime.h>

#define F_DIM 512
#define H_DIM 3
#define C_DIM 7
#define KSTEPS (F_DIM / 4)   // 128 WMMA K-steps of 4

typedef __attribute__((ext_vector_type(2))) float v2f;
typedef __attribute__((ext_vector_type(8))) float v8f;

// ---------------- degree / normalization ----------------

__global__ void gcn_init_deg(float* __restrict__ deg, int n) {
    int i = blockIdx.x * blockDim.x + threadIdx.x;
    if (i < n) deg[i] = 1.0f;  // self-loop contributes 1 to every node's degree
}

__global__ void gcn_edge_deg(const int* __restrict__ dst, float* __restrict__ deg, int e) {
    int i = blockIdx.x * blockDim.x + threadIdx.x;
    if (i < e) atomicAdd(&deg[dst[i]], 1.0f);
}

__global__ void gcn_rsqrt(float* __restrict__ deg, int n) {
    int i = blockIdx.x * blockDim.x + threadIdx.x;
    if (i < n) deg[i] = rsqrtf(deg[i]);   // deg >= 1 always (self loops)
}

// ---------------- xw = x @ W_gcn via V_WMMA_F32_16X16X4_F32 ----------------
// One wave = one 16-row tile of x; K=512 in 128 chunks of 4.
// B fragments (W_gcn zero-padded to 16 cols, pre-swizzled into the WMMA lane
// layout) are staged in LDS once per block: inner loop is
//   global_load_b64 (A, non-temporal) + ds_load_b64 (B) + v_wmma.
// xw is stored with row stride 4 so the scatter phase can gather float4.

__global__ __launch_bounds__(256) void gcn_xw_wmma(const float* __restrict__ x,
                                                   const float* __restrict__ Wg,  // [512][3]
                                                   float* __restrict__ xw,        // [N][4] padded
                                                   int n) {
    __shared__ v2f Blds[KSTEPS * 32];     // 32 KB

    // ---- stage B fragments (before any wave can exit: barrier safety) ----
    for (int idx = threadIdx.x; idx < KSTEPS * 32; idx += blockDim.x) {
        const int l    = idx & 31;
        const int half = l >> 4;
        const int col  = l & 15;
        const int ka   = (idx >> 5) * 4 + 2 * half;
        v2f b;
        b.x = (col < H_DIM) ? Wg[(ka + 0) * H_DIM + col] : 0.f;
        b.y = (col < H_DIM) ? Wg[(ka + 1) * H_DIM + col] : 0.f;
        Blds[idx] = b;
    }
    __syncthreads();

    const int lane   = threadIdx.x & 31;
    const int waveId = threadIdx.x >> 5;
    const int tile   = blockIdx.x * (blockDim.x >> 5) + waveId;
    const int row0   = tile * 16;
    if (row0 >= n) return;                 // wave-uniform: EXEC stays all-1s for WMMA

    const int half = lane >> 4;            // 0: K = k,k+1   1: K = k+2,k+3
    const int m    = lane & 15;            // A-matrix row within tile; also B/C column
    int rowLd = row0 + m;
    if (rowLd >= n) rowLd = n - 1;         // clamp loads (keeps EXEC full); writes guarded
    // per step the lane reads 8B at byte offset (row*512 + step*4 + 2*half)*4
    const v2f* __restrict__ xv =
        (const v2f*)(x + (size_t)rowLd * F_DIM + 2 * half);

    v8f c = {0.f, 0.f, 0.f, 0.f, 0.f, 0.f, 0.f, 0.f};

#pragma unroll 4
    for (int step = 0; step < KSTEPS; ++step) {
        v2f a = __builtin_nontemporal_load(&xv[step * 2]);  // x streamed once: TH=NT
        v2f b = Blds[step * 32 + lane];
        c = __builtin_amdgcn_wmma_f32_16x16x4_f32(false, a, false, b, (short)0, c,
                                                  false, false);
    }

    // C/D layout: VGPR v -> (M = v + 8*half, N = m). Cols 0..2 real, col 3 is zero pad.
    if (m < 4) {
#pragma unroll
        for (int v = 0; v < 8; ++v) {
            const int r = row0 + v + 8 * half;
            if (r < n) xw[(size_t)r * 4 + m] = c[v];
        }
    }
}

// ---------------- aggregation ----------------

__global__ void gcn_init_agg(float* __restrict__ agg, const float* __restrict__ bg, int n3) {
    int i = blockIdx.x * blockDim.x + threadIdx.x;
    if (i < n3) agg[i] = bg[i % H_DIM];    // fold b_gcn into the accumulator init
}

__global__ void gcn_scatter(const int* __restrict__ src, const int* __restrict__ dst,
                            const float* __restrict__ dis, const float4* __restrict__ xw4,
                            float* __restrict__ agg, int e) {
    int i = blockIdx.x * blockDim.x + threadIdx.x;
    if (i >= e) return;
    const int s = src[i];
    const int d = dst[i];
    const float w = dis[s] * dis[d];
    const float4 v = xw4[s];
    atomicAdd(&agg[(size_t)d * 3 + 0], v.x * w);
    atomicAdd(&agg[(size_t)d * 3 + 1], v.y * w);
    atomicAdd(&agg[(size_t)d * 3 + 2], v.z * w);
}

// ---------------- epilogue: self-loop + ReLU + h @ W_out + b_out ----------------

__global__ void gcn_finalize(const float* __restrict__ dis, const float4* __restrict__ xw4,
                             const float* __restrict__ agg, const float* __restrict__ Wo,
                             const float* __restrict__ bo, float* __restrict__ h_out,
                             float* __restrict__ z_out, int n) {
    int i = blockIdx.x * blockDim.x + threadIdx.x;
    if (i >= n) return;
    const float sw = dis[i] * dis[i];      // self-loop norm
    const float4 v = xw4[i];
    float h0 = fmaxf(agg[(size_t)i * 3 + 0] + v.x * sw, 0.f);
    float h1 = fmaxf(agg[(size_t)i * 3 + 1] + v.y * sw, 0.f);
    float h2 = fmaxf(agg[(size_t)i * 3 + 2] + v.z * sw, 0.f);
    h_out[(size_t)i * 3 + 0] = h0;
    h_out[(size_t)i * 3 + 1] = h1;
    h_out[(size_t)i * 3 + 2] = h2;
#pragma unroll
    for (int c = 0; c < C_DIM; ++c) {
        float z = bo[c];
        z = fmaf(h0, Wo[0 * C_DIM + c], z);
        z = fmaf(h1, Wo[1 * C_DIM + c], z);
        z = fmaf(h2, Wo[2 * C_DIM + c], z);
        z_out[(size_t)i * C_DIM + c] = z;
    }
}

// ---------------- launch ----------------

extern "C" void kernel_launch(void* const* d_in, const int* in_sizes, int n_in,
                              void* d_out, int out_size, void* d_ws, size_t ws_size,
                              hipStream_t stream) {
    const float* x     = (const float*)d_in[0];
    const int*   edge  = (const int*)d_in[1];     // [2][E]: row0 = src, row1 = dst
    const float* W_gcn = (const float*)d_in[2];
    const float* b_gcn = (const float*)d_in[3];
    const float* W_out = (const float*)d_in[4];
    const float* b_out = (const float*)d_in[5];

    const int N = in_sizes[0] / F_DIM;
    const int E = in_sizes[1] / 2;
    const int* src = edge;
    const int* dst = edge + E;

    // workspace layout (floats): deg/dis [N] | xw padded [4N] | agg [3N]  => 8N floats
    float* ws  = (float*)d_ws;
    float* deg = ws;                 // reused in-place as dis after gcn_rsqrt
    float* xw  = ws + (size_t)N;
    float* agg = ws + (size_t)5 * N;

    float* h_out = (float*)d_out;                    // [N][3]
    float* z_out = h_out + (size_t)N * H_DIM;        // [N][7]

    const int B = 256;
    gcn_init_deg<<<(N + B - 1) / B, B, 0, stream>>>(deg, N);
    gcn_edge_deg<<<(E + B - 1) / B, B, 0, stream>>>(dst, deg, E);
    gcn_rsqrt<<<(N + B - 1) / B, B, 0, stream>>>(deg, N);

    const int tiles = (N + 15) / 16;
    const int wavesPerBlock = 8;     // 256 threads = 8 wave32
    gcn_xw_wmma<<<(tiles + wavesPerBlock - 1) / wavesPerBlock, 32 * wavesPerBlock, 0, stream>>>(
        x, W_gcn, xw, N);

    gcn_init_agg<<<(3 * N + B - 1) / B, B, 0, stream>>>(agg, b_gcn, 3 * N);
    gcn_scatter<<<(E + B - 1) / B, B, 0, stream>>>(src, dst, deg, (const float4*)xw, agg, E);
    gcn_finalize<<<(N + B - 1) / B, B, 0, stream>>>(deg, (const float4*)xw, agg, W_out, b_out,
                                                    h_out, z_out, N);
}